// DGCNN_19035295055872
// MI455X (gfx1250) — compile-verified
//
#include <hip/hip_runtime.h>
#include <hip/hip_bf16.h>
#include <math.h>

// ---------------- problem constants (from reference) ----------------
#define G_   400
#define NPG_ 128
#define N_   (G_ * NPG_)     // 51200 nodes
#define E_   819200
#define F_   128
#define D_   97              // 32+32+32+1 concatenated features
#define K_   30
#define C1_  16
#define C2_  32
#define KW2_ 5
#define P1LEN_ (K_ / 2)      // 15
#define C2LEN_ (P1LEN_ - KW2_ + 1)  // 11
#define OUTPG_ (C2_ * C2LEN_)       // 352 floats per graph

typedef __attribute__((ext_vector_type(2))) float v2f;
typedef __attribute__((ext_vector_type(8))) float v8f;

// ---------------------------------------------------------------
// Edge scatter-add: agg[dst] += cur[src]   (agg pre-seeded with cur)
// One thread handles a float4 chunk of one edge's feature row:
// global_load_b128 + 4x global_atomic_add_f32. The agg buffer is
// <=26MB -> atomics stay resident in the 192MB L2.
// ---------------------------------------------------------------
template <int FIN>
__global__ void __launch_bounds__(256)
edge_scatter_kernel(const float* __restrict__ cur,
                    const int* __restrict__ esrc,
                    const int* __restrict__ edst,
                    float* __restrict__ agg) {
    constexpr int CPE = FIN / 4;  // float4 chunks per edge
    size_t t = (size_t)blockIdx.x * 256 + threadIdx.x;
    int e = (int)(t / CPE);
    int c = (int)(t % CPE);
    if (e >= E_) return;
    int s = esrc[e];
    int d = edst[e];
    const float4 v = *(const float4*)(cur + (size_t)s * FIN + c * 4);
    float* o = agg + (size_t)d * FIN + c * 4;
    atomicAdd(o + 0, v.x);
    atomicAdd(o + 1, v.y);
    atomicAdd(o + 2, v.z);
    atomicAdd(o + 3, v.w);
}

// ---------------------------------------------------------------
// lin = agg @ W + b ; cur = tanh(lin / deg)  via V_WMMA_F32_16X16X4_F32
// Block = 256 threads = 8 wave32s; each wave owns a 16-row tile and
// produces a 16x32 output (two 16x16 f32 accumulators).
//
// LDS layout tuned so every WMMA operand is ONE aligned ds_load_b64:
//   - W stored TRANSPOSED (sWT[n][k], even pitch FIN+2) so the
//     B-fragment K-pair {W[kg][n], W[kg+1][n]} is contiguous.
//   - sA pitch 36: rows 144B -> 16B-aligned for float4 staging
//     (global_load_b128 + ds_store_b128), still even for aligned b64
//     operand reads; 36*m mod 64 = 4*(9m mod 16) -> 16 distinct banks
//     per lane-half (9 coprime to 16).
//
// ISA VGPR layouts (CDNA5 ISA):
//   A 16x4 f32 : lane m=lane&15; VGPR0=K[2*half], VGPR1=K[2*half+1]
//   B 4x16 f32 : lane n=lane&15; VGPR0=row 2*half, VGPR1=row 2*half+1
//   C/D 16x16  : VGPR v holds M=v (lanes 0-15) / M=v+8 (lanes 16-31)
// ---------------------------------------------------------------
template <int FIN>
__global__ void __launch_bounds__(256)
gnn_gemm_tanh_kernel(const float* __restrict__ agg,
                     const float* __restrict__ degs,
                     const float* __restrict__ W,     // FIN x 32
                     const float* __restrict__ bias,  // 32
                     float* __restrict__ curout,      // N x 32
                     float* __restrict__ cat,         // N x 97
                     int catOff) {
    constexpr int WP = FIN + 2;   // even pitch for transposed W
    __shared__ float sWT[32][WP];   // sWT[n][k] = W[k][n]
    __shared__ float sA[128][36];   // 16B-aligned rows for b128 staging

    const int tid  = threadIdx.x;
    const int wid  = tid >> 5;
    const int lane = tid & 31;
    const int m    = lane & 15;
    const int half = lane >> 4;
    const int n    = lane & 15;
    const int rowBase = blockIdx.x * 128;

    // stage W transposed into LDS
    for (int i = tid; i < FIN * 32; i += 256) {
        const int k  = i >> 5;
        const int nn = i & 31;
        sWT[nn][k] = W[i];
    }

    v8f acc0 = {};
    v8f acc1 = {};

    for (int kc = 0; kc < FIN; kc += 32) {
        __syncthreads();  // protect sA across chunks
        // vectorized staging: 128 rows x 8 float4 chunks = 1024 chunks,
        // 4 per thread: global_load_b128 -> ds_store_b128
#pragma unroll
        for (int i = tid; i < 128 * 8; i += 256) {
            const int r  = i >> 3;
            const int cc = (i & 7) * 4;
            *(float4*)&sA[r][cc] =
                *(const float4*)(agg + (size_t)(rowBase + r) * FIN + kc + cc);
        }
        if (kc + 32 < FIN) {
            // speculative prefetch of the next K-chunk (global_prefetch_b8)
            __builtin_prefetch(agg + (size_t)(rowBase + (tid >> 5) * 16) * FIN + kc + 32, 0, 0);
        }
        __syncthreads();

        const float* aRow  = &sA[wid * 16 + m][half * 2];
        const float* bRow0 = &sWT[n][kc + half * 2];
        const float* bRow1 = &sWT[n + 16][kc + half * 2];
#pragma unroll
        for (int kk = 0; kk < 32; kk += 4) {
            const v2f a  = *(const v2f*)(aRow + kk);   // aligned ds_load_b64
            const v2f b0 = *(const v2f*)(bRow0 + kk);  // aligned ds_load_b64
            const v2f b1 = *(const v2f*)(bRow1 + kk);  // aligned ds_load_b64
            acc0 = __builtin_amdgcn_wmma_f32_16x16x4_f32(false, a, false, b0,
                                                         (short)0, acc0, false, false);
            acc1 = __builtin_amdgcn_wmma_f32_16x16x4_f32(false, a, false, b1,
                                                         (short)0, acc1, false, false);
        }
    }

    // fused epilogue: +bias, /deg, tanh, dual store (dense + concat slab)
    const float bn0 = bias[n];
    const float bn1 = bias[n + 16];
#pragma unroll
    for (int v = 0; v < 8; ++v) {
        const int mr  = v + half * 8;
        const int row = rowBase + wid * 16 + mr;
        const float invd = 1.0f / degs[row];
        const float x0 = tanhf((acc0[v] + bn0) * invd);
        const float x1 = tanhf((acc1[v] + bn1) * invd);
        curout[(size_t)row * 32 + n]          = x0;
        curout[(size_t)row * 32 + n + 16]     = x1;
        cat[(size_t)row * D_ + catOff + n]      = x0;
        cat[(size_t)row * D_ + catOff + n + 16] = x1;
    }
}

// ---------------------------------------------------------------
// Layer 3: out-dim = 1 -> scalar dot per node (not worth WMMA pad)
// ---------------------------------------------------------------
__global__ void __launch_bounds__(256)
layer3_kernel(const float* __restrict__ agg,   // N x 32
              const float* __restrict__ degs,
              const float* __restrict__ W3,    // 32 x 1
              const float* __restrict__ b3,    // 1
              float* __restrict__ cat) {
    int i = blockIdx.x * 256 + threadIdx.x;
    if (i >= N_) return;
    float s = b3[0];
#pragma unroll
    for (int k = 0; k < 32; ++k) s += agg[(size_t)i * 32 + k] * W3[k];
    cat[(size_t)i * D_ + 96] = tanhf(s / degs[i]);
}

// ---------------------------------------------------------------
// Per-graph: stable top-K rank (matches lax.top_k tie-break: lower
// index first), pooled gather, conv1+relu, pair-maxpool, conv2+relu.
// Everything staged in LDS; one 128-thread block per graph.
// ---------------------------------------------------------------
__global__ void __launch_bounds__(128)
pool_conv_kernel(const float* __restrict__ cat,
                 const float* __restrict__ c1w,  // 16 x 97
                 const float* __restrict__ c1b,  // 16
                 const float* __restrict__ c2w,  // 32 x 16 x 5
                 const float* __restrict__ c2b,  // 32
                 float* __restrict__ out,        // G x 352
                 float* __restrict__ topkOut) {  // G x 30 (as float)
    __shared__ float sv[NPG_];
    __shared__ int   sTop[K_];
    __shared__ float sPool[K_][D_];
    __shared__ float sC1[C1_][K_];
    __shared__ float sP1[C1_][P1LEN_];

    const int g   = blockIdx.x;
    const int tid = threadIdx.x;

    sv[tid] = cat[((size_t)g * NPG_ + tid) * D_ + 96];
    __syncthreads();

    // stable rank: count strictly-greater, or equal with smaller index
    const float mine = sv[tid];
    int rank = 0;
#pragma unroll 8
    for (int j = 0; j < NPG_; ++j) {
        const float o = sv[j];
        rank += (o > mine) || (o == mine && j < tid);
    }
    if (rank < K_) {
        sTop[rank] = tid;
        topkOut[(size_t)g * K_ + rank] = (float)tid;
    }
    __syncthreads();

    // gather pooled rows (K x D) into LDS
    for (int i = tid; i < K_ * D_; i += 128) {
        const int k = i / D_;
        const int d = i - k * D_;
        sPool[k][d] = cat[((size_t)g * NPG_ + sTop[k]) * D_ + d];
    }
    __syncthreads();

    // conv1: (16 x 97) . pooled^T -> 16 x 30, relu
    for (int i = tid; i < C1_ * K_; i += 128) {
        const int c = i / K_;
        const int k = i - c * K_;
        float s = c1b[c];
        for (int d = 0; d < D_; ++d) s += c1w[c * D_ + d] * sPool[k][d];
        sC1[c][k] = fmaxf(s, 0.0f);
    }
    __syncthreads();

    // max-pool over adjacent pairs -> 16 x 15
    for (int i = tid; i < C1_ * P1LEN_; i += 128) {
        const int c = i / P1LEN_;
        const int x = i - c * P1LEN_;
        sP1[c][x] = fmaxf(sC1[c][2 * x], sC1[c][2 * x + 1]);
    }
    __syncthreads();

    // conv2 (valid, kw=5) + relu -> 32 x 11, flattened to out
    for (int i = tid; i < C2_ * C2LEN_; i += 128) {
        const int co = i / C2LEN_;
        const int x  = i - co * C2LEN_;
        float s = c2b[co];
#pragma unroll
        for (int ci = 0; ci < C1_; ++ci) {
#pragma unroll
            for (int w = 0; w < KW2_; ++w)
                s += c2w[co * (C1_ * KW2_) + ci * KW2_ + w] * sP1[ci][x + w];
        }
        out[(size_t)g * OUTPG_ + i] = fmaxf(s, 0.0f);
    }
}

// ---------------------------------------------------------------
extern "C" void kernel_launch(void* const* d_in, const int* in_sizes, int n_in,
                              void* d_out, int out_size, void* d_ws, size_t ws_size,
                              hipStream_t stream) {
    const float* node_feat = (const float*)d_in[0];   // N x 128
    const float* degs      = (const float*)d_in[1];   // N x 1
    const int*   esrc      = (const int*)d_in[2];     // E
    const int*   edst      = (const int*)d_in[3];     // E
    const float* W0 = (const float*)d_in[4];
    const float* b0 = (const float*)d_in[5];
    const float* W1 = (const float*)d_in[6];
    const float* b1 = (const float*)d_in[7];
    const float* W2 = (const float*)d_in[8];
    const float* b2 = (const float*)d_in[9];
    const float* W3 = (const float*)d_in[10];
    const float* b3 = (const float*)d_in[11];
    const float* c1w = (const float*)d_in[12];
    const float* c1b = (const float*)d_in[13];
    const float* c2w = (const float*)d_in[14];
    const float* c2b = (const float*)d_in[15];

    float* out     = (float*)d_out;                 // G x 352
    float* topkOut = out + (size_t)G_ * OUTPG_;     // G x 30

    // workspace layout (floats)
    float* ws   = (float*)d_ws;
    float* agg  = ws;                                  // N x 128 (layer0), reused N x 32 after
    float* curA = agg  + (size_t)N_ * F_;              // N x 32
    float* curB = curA + (size_t)N_ * 32;              // N x 32
    float* cat  = curB + (size_t)N_ * 32;              // N x 97

    const int GEMM_BLOCKS = N_ / 128;  // 400

    // ---- layer 0 (FIN = 128) ----
    hipMemcpyAsync(agg, node_feat, (size_t)N_ * F_ * sizeof(float),
                   hipMemcpyDeviceToDevice, stream);
    edge_scatter_kernel<128><<<(E_ * 32) / 256, 256, 0, stream>>>(node_feat, esrc, edst, agg);
    gnn_gemm_tanh_kernel<128><<<GEMM_BLOCKS, 256, 0, stream>>>(agg, degs, W0, b0, curA, cat, 0);

    // ---- layer 1 (FIN = 32) ----
    hipMemcpyAsync(agg, curA, (size_t)N_ * 32 * sizeof(float),
                   hipMemcpyDeviceToDevice, stream);
    edge_scatter_kernel<32><<<(E_ * 8) / 256, 256, 0, stream>>>(curA, esrc, edst, agg);
    gnn_gemm_tanh_kernel<32><<<GEMM_BLOCKS, 256, 0, stream>>>(agg, degs, W1, b1, curB, cat, 32);

    // ---- layer 2 (FIN = 32) ----
    hipMemcpyAsync(agg, curB, (size_t)N_ * 32 * sizeof(float),
                   hipMemcpyDeviceToDevice, stream);
    edge_scatter_kernel<32><<<(E_ * 8) / 256, 256, 0, stream>>>(curB, esrc, edst, agg);
    gnn_gemm_tanh_kernel<32><<<GEMM_BLOCKS, 256, 0, stream>>>(agg, degs, W2, b2, curA, cat, 64);

    // ---- layer 3 (FIN = 32 -> 1) ----
    hipMemcpyAsync(agg, curA, (size_t)N_ * 32 * sizeof(float),
                   hipMemcpyDeviceToDevice, stream);
    edge_scatter_kernel<32><<<(E_ * 8) / 256, 256, 0, stream>>>(curA, esrc, edst, agg);
    layer3_kernel<<<(N_ + 255) / 256, 256, 0, stream>>>(agg, degs, W3, b3, cat);

    // ---- sort-pool top-K + conv1 + maxpool + conv2 ----
    pool_conv_kernel<<<G_, 128, 0, stream>>>(cat, c1w, c1b, c2w, c2b, out, topkOut);
}